// EuclideanCodebook_5068061409925
// MI455X (gfx1250) — compile-verified
//
#include <hip/hip_runtime.h>

typedef __attribute__((ext_vector_type(2))) float v2f;
typedef __attribute__((ext_vector_type(4))) float f4v;
typedef __attribute__((ext_vector_type(8))) float v8f;
typedef __attribute__((ext_vector_type(4))) unsigned int u32x4;
typedef __attribute__((ext_vector_type(8))) int i32x8;

static constexpr int DIMS    = 256;   // D
static constexpr int NCODES  = 8192;  // N
static constexpr int BT      = 8192;  // B*T
static constexpr int M_TILE  = 64;    // tokens per workgroup (4 waves x 16)
static constexpr int THREADS = 128;   // 4 waves of 32
static constexpr int NCHUNK  = 32;    // codes staged in LDS per iteration
static constexpr int NITERS  = NCODES / NCHUNK;   // 256 chunk iterations
static constexpr int KSTEPS  = DIMS / 4;          // 64 WMMA k-steps of depth 4
static constexpr int CHUNK_ELEMS = NCHUNK * DIMS; // 8192 floats = 32 KB
static constexpr int CHUNK_BYTES = CHUNK_ELEMS * 4;

// ---------------------------------------------------------------------------
// LDS byte offset of a shared-memory pointer (generic -> addrspace(3) cast).
// ---------------------------------------------------------------------------
__device__ inline unsigned lds_byte_offset(const void* p) {
  return (unsigned)(uintptr_t)(__attribute__((address_space(3))) const void*)p;
}

// ---------------------------------------------------------------------------
// Issue a TDM load of `nelems` contiguous f32 from global -> LDS.
// D# group0: count=1 | lds_addr | global_addr[56:0] | type=2
// D# group1: data_size=4B, tensor_dim0=tile_dim0=nelems (1 row), dim1=1
// 2-group form (VADDR2/VADDR3 = NULL) for <=2D tensors. Tracked by TENSORcnt.
// ---------------------------------------------------------------------------
__device__ inline void tdm_load_to_lds(unsigned lds_off, const float* gsrc,
                                       unsigned nelems) {
  unsigned long long ga = (unsigned long long)(uintptr_t)gsrc;
  u32x4 g0;
  g0.x = 1u;                                        // count=1, user descriptor
  g0.y = lds_off;                                   // LDS byte address
  g0.z = (unsigned)(ga & 0xffffffffu);              // global_addr[31:0]
  g0.w = (unsigned)((ga >> 32) & 0x01ffffffu)       // global_addr[56:32]
         | 0x80000000u;                             // type = 2 ("image")
  i32x8 g1;
  g1[0] = (int)(2u << 16);                          // data_size = 4 bytes
  g1[1] = (int)(nelems << 16);                      // tensor_dim0 lo16 @ [31:16]
  g1[2] = (int)((nelems >> 16) | (1u << 16));       // dim0 hi16 | tensor_dim1=1
  g1[3] = (int)(nelems << 16);                      // dim1 hi=0 | tile_dim0
  g1[4] = 1;                                        // tile_dim1 = 1
  g1[5] = (int)nelems;                              // tensor_dim0_stride lo32
  g1[6] = 0;                                        // stride hi | dim1_stride lo
  g1[7] = 0;
  asm volatile("tensor_load_to_lds %0, %1" ::"s"(g0), "s"(g1) : "memory");
}

// ---------------------------------------------------------------------------
// Kernel 1: e_sq[n] = ||embed[n]||^2   (8192 values -> d_ws)
// ---------------------------------------------------------------------------
__global__ __launch_bounds__(256) void esq_kernel(const float* __restrict__ embed,
                                                  float* __restrict__ esq) {
  int n = blockIdx.x * blockDim.x + threadIdx.x;
  const f4v* row = (const f4v*)(embed + (size_t)n * DIMS);
  float s = 0.0f;
#pragma unroll 8
  for (int i = 0; i < DIMS / 4; ++i) {
    f4v v = row[i];
    s += v.x * v.x + v.y * v.y + v.z * v.z + v.w * v.w;
  }
  esq[n] = s;
}

// ---------------------------------------------------------------------------
// Kernel 2: fused encode (fp32 WMMA distance matmul + argmin) and decode
// (gather). 4 waves x 16 tokens; A-fragments register-resident; embed chunks
// double-buffered in LDS via the Tensor Data Mover.
// ---------------------------------------------------------------------------
__global__ __launch_bounds__(THREADS) void vq_kernel(const float* __restrict__ x,
                                                     const float* __restrict__ embed,
                                                     const float* __restrict__ esq,
                                                     float* __restrict__ quant_out,
                                                     int* __restrict__ ind_out) {
  __shared__ float smem[2 * CHUNK_ELEMS];  // 64 KB: two 32 KB embed buffers
  int* ind_lds = (int*)smem;               // storage reused after compute loop

  const int tid  = threadIdx.x;
  const int lane = tid & 31;
  const int wave = tid >> 5;
  const int col  = lane & 15;   // N column (B/C/D) or M row (A) within tile
  const int hi   = lane >> 4;   // which 16-lane half
  const int koff = hi * 2;      // K sub-offset for 16x16x4 fp32 layout

  const int tokenBase = blockIdx.x * M_TILE;
  const int tok = tokenBase + wave * 16 + col;  // this lane's A-matrix row

  const unsigned lds_base = lds_byte_offset(smem);

  // ---- Load A fragments (x rows) once into registers, WMMA lane layout:
  // lanes 0-15: K = 4k, 4k+1 ; lanes 16-31: K = 4k+2, 4k+3 (row M = col).
  const float* xrow = x + (size_t)tok * DIMS;
  v2f afrag[KSTEPS];
#pragma unroll
  for (int k = 0; k < KSTEPS; ++k) {
    afrag[k] = *(const v2f*)(xrow + 4 * k + koff);
  }

  // Per-lane running argmin for 8 accumulator rows (C/D VGPR j -> row j+8*hi).
  float minval[8];
  int minidx[8];
#pragma unroll
  for (int j = 0; j < 8; ++j) {
    minval[j] = 3.4e38f;
    minidx[j] = 0;
  }

  // ---- Prologue: TDM chunk 0 into buffer 0.
  if (wave == 0) {
    tdm_load_to_lds(lds_base, embed, CHUNK_ELEMS);
    __builtin_amdgcn_s_wait_tensorcnt(0);
  }
  __syncthreads();

  for (int i = 0; i < NITERS; ++i) {
    const int nbase = i * NCHUNK;
    const float* ebuf = smem + (size_t)(i & 1) * CHUNK_ELEMS;

    // Kick off TDM for the next chunk into the other buffer (overlaps WMMA).
    if (wave == 0 && i + 1 < NITERS) {
      tdm_load_to_lds(lds_base + ((i + 1) & 1) * CHUNK_BYTES,
                      embed + (size_t)(i + 1) * CHUNK_ELEMS, CHUNK_ELEMS);
    }

    // Two 16-code tiles per chunk; both share the A fragment.
    v8f acc0 = {};
    v8f acc1 = {};
    const float* b0p = ebuf + (size_t)col * DIMS;         // code col
    const float* b1p = ebuf + (size_t)(col + 16) * DIMS;  // code col+16
#pragma unroll
    for (int k = 0; k < KSTEPS; ++k) {
      v2f b0 = *(const v2f*)(b0p + 4 * k + koff);
      v2f b1 = *(const v2f*)(b1p + 4 * k + koff);
      acc0 = __builtin_amdgcn_wmma_f32_16x16x4_f32(false, afrag[k], false, b0,
                                                   (short)0, acc0, false, false);
      acc1 = __builtin_amdgcn_wmma_f32_16x16x4_f32(false, afrag[k], false, b1,
                                                   (short)0, acc1, false, false);
    }

    // dist = ||e||^2 - 2 x.e  (x_sq constant per token -> argmin-equivalent)
    const float e0 = esq[nbase + col];
    const float e1 = esq[nbase + 16 + col];
    const int n0 = nbase + col;
    const int n1 = nbase + 16 + col;
#pragma unroll
    for (int j = 0; j < 8; ++j) {
      float d0 = e0 - 2.0f * acc0[j];
      if (d0 < minval[j]) { minval[j] = d0; minidx[j] = n0; }
      float d1 = e1 - 2.0f * acc1[j];
      if (d1 < minval[j]) { minval[j] = d1; minidx[j] = n1; }
    }

    // Drain the in-flight TDM, then release both buffers to the workgroup.
    if (wave == 0 && i + 1 < NITERS) {
      __builtin_amdgcn_s_wait_tensorcnt(0);
    }
    __syncthreads();
  }

  // ---- Cross-lane argmin within each 16-lane half (N is across lanes 0-15).
#pragma unroll
  for (int j = 0; j < 8; ++j) {
    float v = minval[j];
    int ix = minidx[j];
#pragma unroll
    for (int off = 1; off < 16; off <<= 1) {
      float ov = __shfl_xor(v, off, 32);
      int oi = __shfl_xor(ix, off, 32);
      if (ov < v || (ov == v && oi < ix)) { v = ov; ix = oi; }
    }
    if (col == 0) {
      int m = j + hi * 8;            // row within this wave's 16-token tile
      int t = wave * 16 + m;         // token within workgroup
      ind_lds[t] = ix;
      ind_out[tokenBase + t] = ix;
    }
  }
  __syncthreads();

  // ---- Decode: quantized[t] = embed[ind[t]]  (2 threads per token).
  {
    const int tloc = tid >> 1;
    const int h = tid & 1;
    const int code = ind_lds[tloc];
    const f4v* srcr = (const f4v*)(embed + (size_t)code * DIMS) + h * 32;
    f4v* dstr = (f4v*)(quant_out + (size_t)(tokenBase + tloc) * DIMS) + h * 32;
#pragma unroll
    for (int i = 0; i < 32; ++i) dstr[i] = srcr[i];
  }
}

// ---------------------------------------------------------------------------
extern "C" void kernel_launch(void* const* d_in, const int* in_sizes, int n_in,
                              void* d_out, int out_size, void* d_ws, size_t ws_size,
                              hipStream_t stream) {
  const float* x     = (const float*)d_in[0];  // [8,1024,256] f32
  const float* embed = (const float*)d_in[1];  // [8192,256]   f32

  float* quant_out = (float*)d_out;                              // BT*DIMS f32
  int*   ind_out   = (int*)((float*)d_out + (size_t)BT * DIMS);  // BT int32
  float* esq       = (float*)d_ws;                               // 8192 f32

  esq_kernel<<<NCODES / 256, 256, 0, stream>>>(embed, esq);
  vq_kernel<<<BT / M_TILE, THREADS, 0, stream>>>(x, embed, esq, quant_out, ind_out);
}